// MoE_78503412236862
// MI455X (gfx1250) — compile-verified
//
#include <hip/hip_runtime.h>
#include <math.h>

// ---------------- problem constants ----------------
#define D_    1024
#define H_    1365
#define E_    7
#define N_    2048
#define TOPK  2

// ---------------- tiling ----------------
#define HC        256        // H-chunk per outer loop (8 waves x 2 tiles x 16 cols)
#define NFULL     5          // 5 full chunks (1280 cols) + 1 tail chunk (85 cols)
#define DC        512        // D-chunk staged in LDS for GEMM1
#define XS_STRIDE 516        // 512 + 4 pad  -> conflict-free, rows stay 16B aligned
#define HS_STRIDE 260        // 256 + 4 pad

typedef __attribute__((ext_vector_type(2))) float v2f;
typedef __attribute__((ext_vector_type(8))) float v8f;

#define V8F_ZERO ((v8f){0.f,0.f,0.f,0.f,0.f,0.f,0.f,0.f})

__device__ __forceinline__ float gelu_exact(float v) {
    return 0.5f * v * (1.0f + erff(v * 0.70710678118654752440f));
}

// ---------------- kernel 0: zero output + scratch ----------------
__global__ void moe_init(float* out, int* cnt, float* gate_sum, float* z2) {
    int gid = blockIdx.x * blockDim.x + threadIdx.x;
    if (gid < N_ * D_) out[gid] = 0.0f;
    if (gid < E_) { cnt[gid] = 0; gate_sum[gid] = 0.0f; }
    if (gid == 0) z2[0] = 0.0f;
}

// ---------------- kernel 1: router (one wave32 per token) ----------------
__global__ __launch_bounds__(128) void moe_router(
    const float* __restrict__ x, const float* __restrict__ Wg,
    int* cnt, float* gate_sum, float* z2, int* toklist, float* gwlist)
{
    const int lane = threadIdx.x & 31;
    const int wv   = threadIdx.x >> 5;
    const int n    = blockIdx.x * 4 + wv;
    if (n >= N_) return;

    float acc[E_];
#pragma unroll
    for (int e = 0; e < E_; ++e) acc[e] = 0.0f;
    const float* xr = x + (size_t)n * D_;
    for (int d = lane; d < D_; d += 32) {
        float xv = xr[d];
#pragma unroll
        for (int e = 0; e < E_; ++e) acc[e] = fmaf(xv, Wg[e * D_ + d], acc[e]);
    }
#pragma unroll
    for (int e = 0; e < E_; ++e) {
#pragma unroll
        for (int off = 16; off > 0; off >>= 1)
            acc[e] += __shfl_xor(acc[e], off, 32);
    }
    if (lane == 0) {
        float mx = acc[0];
#pragma unroll
        for (int e = 1; e < E_; ++e) mx = fmaxf(mx, acc[e]);
        float p[E_]; float se = 0.0f;
#pragma unroll
        for (int e = 0; e < E_; ++e) { p[e] = __expf(acc[e] - mx); se += p[e]; }
        float inv = 1.0f / se;
#pragma unroll
        for (int e = 0; e < E_; ++e) atomicAdd(&gate_sum[e], p[e] * inv);
        float lse = mx + __logf(se);
        atomicAdd(z2, lse * lse);

        int i0 = 0;
#pragma unroll
        for (int e = 1; e < E_; ++e) if (acc[e] > acc[i0]) i0 = e;
        int i1 = (i0 == 0) ? 1 : 0;
#pragma unroll
        for (int e = 0; e < E_; ++e) if (e != i0 && acc[e] > acc[i1]) i1 = e;
        float m2 = fmaxf(acc[i0], acc[i1]);
        float e0 = __expf(acc[i0] - m2), e1 = __expf(acc[i1] - m2);
        float w0 = e0 / (e0 + e1), w1 = e1 / (e0 + e1);

        int p0 = atomicAdd(&cnt[i0], 1);
        toklist[i0 * N_ + p0] = n; gwlist[i0 * N_ + p0] = w0;
        int p1 = atomicAdd(&cnt[i1], 1);
        toklist[i1 * N_ + p1] = n; gwlist[i1 * N_ + p1] = w1;
    }
}

// ---------------- fused FFN chunk processor ----------------
// TAIL=false: all 256 h-cols valid -> branch-free inner loops.
// TAIL=true : clamp addresses; hs pad cols forced to exact 0 so clamped
//             (finite, in-bounds) B values are annihilated in GEMM2.
template<bool TAIL>
__device__ __forceinline__ void ffn_chunk(
    int hcb, int mrow, int koff, int half, int wv,
    const float* __restrict__ xrow,      // this thread's staging row in x
    int srow, int scol,                  // staging row (tid>>4), col base ((tid&15)*4)
    const float* __restrict__ pW1, const float* __restrict__ pB1,
    const float* __restrict__ pW2,
    float* __restrict__ xs, float* __restrict__ hs,
    v8f accy[8])
{
    const int hg0   = hcb * HC;
    const int hcol0 = hg0 + wv * 32 + mrow;        // this wave's two h-tiles
    const int hcol1 = hcol0 + 16;
    const int hc0c  = TAIL ? min(hcol0, H_ - 1) : hcol0;
    const int hc1c  = TAIL ? min(hcol1, H_ - 1) : hcol1;
    const float* w1r0 = pW1 + (size_t)hc0c * D_ + koff;
    const float* w1r1 = pW1 + (size_t)hc1c * D_ + koff;

    v8f acch0a = V8F_ZERO, acch0b = V8F_ZERO;      // even/odd-K chains per tile
    v8f acch1a = V8F_ZERO, acch1b = V8F_ZERO;

    // ---- GEMM1: h[16 x 32] += x[16 x D] * W1^T, K split into DC chunks ----
    for (int dcb = 0; dcb < D_; dcb += DC) {
        __syncthreads();                     // xs / hs free to overwrite
        {
            const float* src = xrow + dcb + scol;
            float*       dst = xs + srow * XS_STRIDE + scol;
#pragma unroll
            for (int it = 0; it < DC / 64; ++it)       // 8 x b128 per thread
                *(float4*)(dst + it * 64) = *(const float4*)(src + it * 64);
        }
        __syncthreads();
        const float* wa = w1r0 + dcb;
        const float* wb = w1r1 + dcb;
#pragma unroll 2
        for (int k = 0; k < DC; k += 8) {
            v2f a0, a1;
            a0.x = xs[mrow * XS_STRIDE + k + koff];
            a0.y = xs[mrow * XS_STRIDE + k + koff + 1];
            a1.x = xs[mrow * XS_STRIDE + k + 4 + koff];
            a1.y = xs[mrow * XS_STRIDE + k + 4 + koff + 1];
            v2f b00; b00.x = wa[k];     b00.y = wa[k + 1];
            v2f b10; b10.x = wb[k];     b10.y = wb[k + 1];
            v2f b01; b01.x = wa[k + 4]; b01.y = wa[k + 5];
            v2f b11; b11.x = wb[k + 4]; b11.y = wb[k + 5];
            acch0a = __builtin_amdgcn_wmma_f32_16x16x4_f32(
                         false, a0, false, b00, (short)0, acch0a, false, false);
            acch1a = __builtin_amdgcn_wmma_f32_16x16x4_f32(
                         false, a0, false, b10, (short)0, acch1a, false, false);
            acch0b = __builtin_amdgcn_wmma_f32_16x16x4_f32(
                         false, a1, false, b01, (short)0, acch0b, false, false);
            acch1b = __builtin_amdgcn_wmma_f32_16x16x4_f32(
                         false, a1, false, b11, (short)0, acch1b, false, false);
        }
    }

    // ---- bias + exact GELU -> LDS (pad cols forced to 0) ----
    const float bb0 = pB1[hc0c];
    const float bb1 = pB1[hc1c];
    __syncthreads();
#pragma unroll
    for (int r = 0; r < 8; ++r) {
        int m = r + half;
        float g0 = gelu_exact(acch0a[r] + acch0b[r] + bb0);
        float g1 = gelu_exact(acch1a[r] + acch1b[r] + bb1);
        if (TAIL) {
            if (hcol0 >= H_) g0 = 0.0f;
            if (hcol1 >= H_) g1 = 0.0f;
        }
        hs[m * HS_STRIDE + wv * 32 + mrow]      = g0;
        hs[m * HS_STRIDE + wv * 32 + 16 + mrow] = g1;
    }
    __syncthreads();

    // ---- GEMM2: k outer / tile inner -> A reused by 8 independent chains ----
    const float* w2base = pW2 + (size_t)(wv * 128 + mrow) * H_ + hg0;
#pragma unroll 2
    for (int k = 0; k < HC; k += 4) {
        int off = k + koff;
        if (TAIL) off = min(off, H_ - 2 - hg0);   // clamped; A cols there are 0
        v2f a;
        a.x = hs[mrow * HS_STRIDE + k + koff];
        a.y = hs[mrow * HS_STRIDE + k + koff + 1];
#pragma unroll
        for (int t = 0; t < 8; ++t) {
            const float* wp = w2base + (size_t)t * 16 * H_ + off;
            v2f b; b.x = wp[0]; b.y = wp[1];
            accy[t] = __builtin_amdgcn_wmma_f32_16x16x4_f32(
                          false, a, false, b, (short)0, accy[t], false, false);
        }
    }
}

// ---------------- kernel 2: fused expert FFN via f32 WMMA ----------------
// grid = (128 token-tiles, E_+1); blockIdx.y == E_ => shared expert.
// block = 256 threads = 8 waves; wave w owns output D-cols [w*128, w*128+128).
__global__ __launch_bounds__(256) void moe_ffn(
    const float* __restrict__ x,
    const float* __restrict__ W1, const float* __restrict__ b1,
    const float* __restrict__ W2, const float* __restrict__ b2,
    const float* __restrict__ Ws1, const float* __restrict__ bs1,
    const float* __restrict__ Ws2, const float* __restrict__ bs2,
    const int* __restrict__ cnt, const int* __restrict__ toklist,
    const float* __restrict__ gwlist, float* __restrict__ out)
{
    __shared__ float xs[16 * XS_STRIDE];   // 33 KB
    __shared__ float hs[16 * HS_STRIDE];   // 16.6 KB
    __shared__ int   st_tok[16];
    __shared__ float st_gw[16];

    const int tile = blockIdx.x;
    const int e    = blockIdx.y;
    const bool is_shared = (e == E_);
    const int m_cnt = is_shared ? N_ : cnt[e];
    if (tile * 16 >= m_cnt) return;

    const float* pW1 = is_shared ? Ws1 : (W1 + (size_t)e * H_ * D_);
    const float* pB1 = is_shared ? bs1 : (b1 + (size_t)e * H_);
    const float* pW2 = is_shared ? Ws2 : (W2 + (size_t)e * D_ * H_);
    const float* pB2 = is_shared ? bs2 : (b2 + (size_t)e * D_);

    if (threadIdx.x < 16) {
        int idx = tile * 16 + threadIdx.x;
        if (is_shared)        { st_tok[threadIdx.x] = idx;                 st_gw[threadIdx.x] = 1.0f; }
        else if (idx < m_cnt) { st_tok[threadIdx.x] = toklist[e*N_ + idx]; st_gw[threadIdx.x] = gwlist[e*N_ + idx]; }
        else                  { st_tok[threadIdx.x] = toklist[e*N_];       st_gw[threadIdx.x] = 0.0f; }
    }
    __syncthreads();

    const int lane = threadIdx.x & 31;
    const int wv   = threadIdx.x >> 5;
    const int mrow = lane & 15;
    const int koff = (lane >> 4) * 2;
    const int half = (lane >> 4) << 3;

    // staging assignment: fixed row per thread; token read from LDS exactly once
    const int srow = threadIdx.x >> 4;          // 0..15
    const int scol = (threadIdx.x & 15) << 2;   // 0,4,...,60
    const float* xrow = x + (size_t)st_tok[srow] * D_;

    v8f accy[8];
#pragma unroll
    for (int t = 0; t < 8; ++t) accy[t] = V8F_ZERO;

    for (int hcb = 0; hcb < NFULL; ++hcb)
        ffn_chunk<false>(hcb, mrow, koff, half, wv, xrow, srow, scol,
                         pW1, pB1, pW2, xs, hs, accy);
    ffn_chunk<true>(NFULL, mrow, koff, half, wv, xrow, srow, scol,
                    pW1, pB1, pW2, xs, hs, accy);

    // ---- writeout: gate-scale (+b2) and accumulate into out ----
#pragma unroll
    for (int t = 0; t < 8; ++t) {
        const int dcol = wv * 128 + t * 16 + mrow;
        const float b2v = pB2[dcol];
#pragma unroll
        for (int r = 0; r < 8; ++r) {
            int m = r + half;
            float val = (accy[t][r] + b2v) * st_gw[m];
            atomicAdd(&out[(size_t)st_tok[m] * D_ + dcol], val);
        }
    }
}

// ---------------- kernel 3: aux losses ----------------
__global__ void moe_aux(const int* cnt, const float* gate_sum, const float* z2, float* out) {
    if (blockIdx.x == 0 && threadIdx.x == 0) {
        float la = 0.0f;
        for (int e = 0; e < E_; ++e)
            la += ((float)cnt[e] / (float)(N_ * TOPK)) * (gate_sum[e] / (float)N_);
        out[(size_t)N_ * D_]     = 0.01f * (float)E_ * la;
        out[(size_t)N_ * D_ + 1] = 0.001f * (z2[0] / (float)N_);
    }
}

// ---------------- host launcher ----------------
extern "C" void kernel_launch(void* const* d_in, const int* in_sizes, int n_in,
                              void* d_out, int out_size, void* d_ws, size_t ws_size,
                              hipStream_t stream)
{
    (void)in_sizes; (void)n_in; (void)out_size; (void)ws_size;
    const float* x   = (const float*)d_in[0];
    const float* Wg  = (const float*)d_in[1];
    const float* W1  = (const float*)d_in[2];
    const float* b1  = (const float*)d_in[3];
    const float* W2  = (const float*)d_in[4];
    const float* b2  = (const float*)d_in[5];
    const float* Ws1 = (const float*)d_in[6];
    const float* bs1 = (const float*)d_in[7];
    const float* Ws2 = (const float*)d_in[8];
    const float* bs2 = (const float*)d_in[9];
    float* out = (float*)d_out;

    char* ws = (char*)d_ws;
    int*   cnt      = (int*)(ws);                  // 8 ints
    float* gate_sum = (float*)(ws + 64);           // 8 floats
    float* z2       = (float*)(ws + 128);          // 1 float
    int*   toklist  = (int*)(ws + 256);            // E*N ints
    float* gwlist   = (float*)(ws + 256 + (size_t)E_ * N_ * sizeof(int));

    moe_init<<<(N_ * D_ + 255) / 256, 256, 0, stream>>>(out, cnt, gate_sum, z2);
    moe_router<<<N_ / 4, 128, 0, stream>>>(x, Wg, cnt, gate_sum, z2, toklist, gwlist);
    dim3 g(N_ / 16, E_ + 1);
    moe_ffn<<<g, 256, 0, stream>>>(x, W1, b1, W2, b2, Ws1, bs1, Ws2, bs2,
                                   cnt, toklist, gwlist, out);
    moe_aux<<<1, 32, 0, stream>>>(cnt, gate_sum, z2, out);
}